// GAFFA_32615981646106
// MI455X (gfx1250) — compile-verified
//
#include <hip/hip_runtime.h>
#include <cstdint>
#include <cstddef>

// ---------------- problem constants ----------------
#define L_N  37
#define K_N  8
#define HM_N 512
#define HS_N 128
#define HC_N 64
#define DELTA_F 1e-6f
#define BN_EPS_F 1e-5f

typedef __attribute__((ext_vector_type(2))) float v2f;
typedef __attribute__((ext_vector_type(8))) float v8f;
typedef int v4i __attribute__((vector_size(16)));

// ---------------- CDNA5 async global->LDS helpers ----------------
#if __has_builtin(__builtin_amdgcn_global_load_async_to_lds_b128)
#define HAS_ASYNC_LDS 1
#else
#define HAS_ASYNC_LDS 0
#endif

typedef __attribute__((address_space(1))) v4i* as1_v4i_p;
typedef __attribute__((address_space(3))) v4i* as3_v4i_p;

__device__ __forceinline__ void g2lds16(const float* g, float* l) {
#if HAS_ASYNC_LDS
  as1_v4i_p gp = reinterpret_cast<as1_v4i_p>(reinterpret_cast<uintptr_t>(g));
  as3_v4i_p lp = reinterpret_cast<as3_v4i_p>(
      static_cast<uint32_t>(reinterpret_cast<uintptr_t>(l)));
  __builtin_amdgcn_global_load_async_to_lds_b128(gp, lp, 0, 0);
#else
  *reinterpret_cast<float4*>(l) = *reinterpret_cast<const float4*>(g);
#endif
}

__device__ __forceinline__ void zero_lds16(float* l) {
  float4 z = make_float4(0.f, 0.f, 0.f, 0.f);
  *reinterpret_cast<float4*>(l) = z;
}

__device__ __forceinline__ void wait_async_lds() {
#if HAS_ASYNC_LDS
#if __has_builtin(__builtin_amdgcn_s_wait_asynccnt)
  __builtin_amdgcn_s_wait_asynccnt(0);
#else
  asm volatile("s_wait_asynccnt 0" ::: "memory");
#endif
#endif
}

// ---------------- math helpers ----------------
__device__ __forceinline__ float sp_b(float x) {  // softplus(5x)/5
  float z = 5.f * x;
  if (z > 20.f) return x;
  return 0.2f * log1pf(expf(z));
}

__device__ __forceinline__ float cubic_w(float x) {  // Keys a=-0.5
  x = fabsf(x);
  if (x < 1.f) return ((1.5f * x - 2.5f) * x) * x + 1.f;
  if (x < 2.f) return ((-0.5f * x + 2.5f) * x - 4.f) * x + 2.f;
  return 0.f;
}

// ---------------- 1: BN stats per layer ----------------
__global__ __launch_bounds__(256) void k_bn_stats(const float* __restrict__ hm,
                                                  const float* __restrict__ gamma,
                                                  float* __restrict__ st_mean,
                                                  float* __restrict__ st_scale) {
  const int l = blockIdx.x, tid = threadIdx.x;
  __shared__ float sa[256], sq[256];
  float a = 0.f, q = 0.f;
  for (int n = tid; n < 2 * HM_N * HM_N; n += 256) {
    int b = n >> 18, pos = n & 262143;
    float v = hm[(size_t)(b * L_N + l) * (HM_N * HM_N) + pos];
    a += v; q += v * v;
  }
  sa[tid] = a; sq[tid] = q; __syncthreads();
  for (int st = 128; st; st >>= 1) {
    if (tid < st) { sa[tid] += sa[tid + st]; sq[tid] += sq[tid + st]; }
    __syncthreads();
  }
  if (tid == 0) {
    const float inv = 1.f / (float)(2 * HM_N * HM_N);
    float m = sa[0] * inv;
    float var = sq[0] * inv - m * m;
    st_mean[l] = m;
    st_scale[l] = gamma[l] * rsqrtf(var + BN_EPS_F);
  }
}

// ---------------- 2: invert valid_idx: per (image j, k) find l ----------------
__global__ void k_pairs(const int* __restrict__ vidx, int* __restrict__ pair_l) {
  int t = threadIdx.x;
  if (t >= L_N * K_N) return;
  int j = t >> 3, k = t & 7;
  int found = 0;
  for (int l = 0; l < L_N; ++l)
    if (vidx[l * K_N + k] == j) { found = l; break; }
  pair_l[t] = found;
}

// ---------------- 3: antialiased cubic downsample 512->128, rows pass ----------------
// tmp1[b][l][y(512)][j(128)] : 16-tap along x, weights cubic((t-7.5)/4), edge renorm
__global__ __launch_bounds__(256) void k_resize_rows(const float* __restrict__ hm,
                                                     float* __restrict__ tmp1) {
  size_t t = (size_t)blockIdx.x * 256 + threadIdx.x;
  if (t >= (size_t)2 * L_N * HM_N * HS_N) return;
  int j = t & 127;
  int y = (t >> 7) & 511;
  int bl = (int)(t >> 16);
  const float* row = hm + (size_t)bl * (HM_N * HM_N) + (size_t)y * HM_N;
  int base = 4 * j - 6;
  float s = 0.f, ws = 0.f;
#pragma unroll
  for (int tt = 0; tt < 16; ++tt) {
    int x = base + tt;
    if (x >= 0 && x < HM_N) {
      float w = cubic_w(((float)tt - 7.5f) * 0.25f);
      s += w * row[x]; ws += w;
    }
  }
  tmp1[t] = s / ws;
}

// ---------------- 4: cols pass + BN affine + softplus, write zero-padded Lpad ----
// Lpad[b][l][256][256]; data at [63..190]^2, else 0
__global__ __launch_bounds__(256) void k_resize_cols_pad(const float* __restrict__ tmp1,
                                                         const float* __restrict__ st_mean,
                                                         const float* __restrict__ st_scale,
                                                         const float* __restrict__ beta,
                                                         float* __restrict__ Lpad) {
  size_t t = (size_t)blockIdx.x * 256 + threadIdx.x;
  if (t >= (size_t)2 * L_N * 256 * 256) return;
  int c = t & 255;
  int a = (t >> 8) & 255;
  int bl = (int)(t >> 16);
  int l = bl % L_N;
  float v = 0.f;
  if (a >= 63 && a <= 190 && c >= 63 && c <= 190) {
    int i = a - 63, j = c - 63;
    int base = 4 * i - 6;
    float s = 0.f, ws = 0.f;
#pragma unroll
    for (int tt = 0; tt < 16; ++tt) {
      int y = base + tt;
      if (y >= 0 && y < HM_N) {
        float w = cubic_w(((float)tt - 7.5f) * 0.25f);
        s += w * tmp1[((size_t)bl * HM_N + y) * HS_N + j]; ws += w;
      }
    }
    float r = s / ws;
    float hn = (r - st_mean[l]) * st_scale[l] + beta[l];
    v = sp_b(hn);
  }
  Lpad[t] = v;
}

// ---------------- 5: prior = softplus(cubic upsample 64->128) ----------------
__global__ __launch_bounds__(256) void k_prior(const float* __restrict__ cond,
                                               float* __restrict__ prior) {
  size_t t = (size_t)blockIdx.x * 256 + threadIdx.x;
  if (t >= (size_t)L_N * K_N * HS_N * HS_N) return;
  int j = t & 127;
  int i = (t >> 7) & 127;
  int lk = (int)(t >> 14);
  const float* src = cond + (size_t)lk * (HC_N * HC_N);
  float yf = (float)i * 0.5f - 0.25f;
  float xf = (float)j * 0.5f - 0.25f;
  int iy = (int)floorf(yf), ix = (int)floorf(xf);
  float wy[4], wx[4], sy = 0.f, sx = 0.f;
#pragma unroll
  for (int u = 0; u < 4; ++u) {
    int ty = iy - 1 + u;
    float w = (ty >= 0 && ty < HC_N) ? cubic_w(yf - (float)ty) : 0.f;
    wy[u] = w; sy += w;
    int tx = ix - 1 + u;
    float wc = (tx >= 0 && tx < HC_N) ? cubic_w(xf - (float)tx) : 0.f;
    wx[u] = wc; sx += wc;
  }
  float acc = 0.f;
#pragma unroll
  for (int u = 0; u < 4; ++u) {
    if (wy[u] == 0.f) continue;
    const float* r = src + (iy - 1 + u) * HC_N;
#pragma unroll
    for (int w = 0; w < 4; ++w)
      if (wx[w] != 0.f) acc += wy[u] * wx[w] * r[ix - 1 + w];
  }
  prior[t] = sp_b(acc / (sy * sx));
}

// ---------------- 6: zero the 128x128 energy accumulator ----------------
__global__ __launch_bounds__(256) void k_zero(float* __restrict__ p, size_t n) {
  size_t t = (size_t)blockIdx.x * 256 + threadIdx.x;
  if (t < n) p[t] = 0.f;
}

// ---------------- 7: WMMA direct conv via shift decomposition ----------------
// O[i,j] = sum_d sum_v P[i-d,v] * Lpad[127+d, 127+j-v]
// grid (8 i-tiles, 37 images, 2 batch); 8 waves, wave w owns j-tile w (16 cols).
#define LSTRIDE 264
#define PSTRIDE 132
__global__ __launch_bounds__(256, 2) void k_conv_wmma(const float* __restrict__ Lpad,
                                                      const float* __restrict__ prior,
                                                      const float* __restrict__ bias,
                                                      const int* __restrict__ pair_l,
                                                      float* __restrict__ energy128) {
  __shared__ float ldsL[16 * LSTRIDE];  // 16 Lpad rows of current d-chunk
  __shared__ float ldsP[32 * PSTRIDE];  // 31-row prior window (row 31 = zero pad)

  const int i0 = blockIdx.x << 4;
  const int jimg = blockIdx.y;
  const int b = blockIdx.z;
  const int tid = threadIdx.x;
  const int lane = tid & 31;
  const int half = lane >> 4;   // 0|1
  const int lrow = lane & 15;   // A:M index, B/D:N index
  const int jc0 = (tid >> 5) << 4;
  const int kA = half << 1;     // K lane split per 16x16x4 f32 A/B layout

  const float* Lbase = Lpad + (size_t)(b * L_N + jimg) * 256 * 256;

  v8f acc[K_N];
  const v8f z8 = {0.f, 0.f, 0.f, 0.f, 0.f, 0.f, 0.f, 0.f};
#pragma unroll
  for (int p = 0; p < K_N; ++p) acc[p] = z8;

  const int dlo = i0 - 127, dhi = i0 + 15;
  for (int dc = dlo; dc <= dhi; dc += 16) {
    __syncthreads();  // previous chunk fully consumed
    // stage 16 Lpad rows: a = 127+dc+r in [0,255] always
    for (int idx = tid; idx < 16 * 64; idx += 256) {
      int r = idx >> 6, c4 = (idx & 63) << 2;
      int a = 127 + dc + r;
      g2lds16(Lbase + (size_t)a * 256 + c4, &ldsL[r * LSTRIDE + c4]);
    }
    // prefetch next chunk's rows through L2
    {
      int an = 127 + dc + 16 + tid;
      if (tid < 16 && an < 256) __builtin_prefetch(Lbase + (size_t)an * 256, 0, 0);
    }
    const int rowP0 = i0 - dc - 15;
#pragma unroll
    for (int p = 0; p < K_N; ++p) {
      const int l = pair_l[jimg * K_N + p];
      const float* Pbase = prior + (size_t)(l * K_N + p) * (HS_N * HS_N);
      __syncthreads();  // previous pair's P window consumed
      for (int idx = tid; idx < 32 * 32; idx += 256) {
        int w = idx >> 5, c4 = (idx & 31) << 2;
        int row = rowP0 + w;
        if (row >= 0 && row < HS_N && w < 31)
          g2lds16(Pbase + (size_t)row * HS_N + c4, &ldsP[w * PSTRIDE + c4]);
        else
          zero_lds16(&ldsP[w * PSTRIDE + c4]);
      }
      wait_async_lds();
      __syncthreads();

      const int cb = 127 + jc0 + lrow - kA;  // B column base (before -vb)
      const int wPb = lrow + 15;             // P window row base (before -dd)
#pragma unroll 1
      for (int dd = 0; dd < 16; ++dd) {
        if (dc + dd > dhi) break;
        const float* Lr = &ldsL[dd * LSTRIDE];
        const float* Pr = &ldsP[(wPb - dd) * PSTRIDE];
#pragma unroll 4
        for (int vb = 0; vb < HS_N; vb += 4) {
          v2f A, B;
          A.x = Pr[vb + kA];
          A.y = Pr[vb + kA + 1];
          B.y = Lr[cb - vb - 1];
          B.x = Lr[cb - vb];
          acc[p] = __builtin_amdgcn_wmma_f32_16x16x4_f32(
              false, A, false, B, (short)0, acc[p], false, false);
        }
      }
    }
  }

  // finalize: energy[b,l,i,j] += log(conv + softplus(bias) + delta)
#pragma unroll
  for (int p = 0; p < K_N; ++p) {
    const int l = pair_l[jimg * K_N + p];
    const float* Bb = bias + (size_t)(l * K_N + p) * (HS_N * HS_N);
    float* Eb = energy128 + (size_t)(b * L_N + l) * (HS_N * HS_N);
#pragma unroll
    for (int e = 0; e < 8; ++e) {
      int i = i0 + e + (half << 3);
      int j = jc0 + lrow;
      float cv = acc[p][e];
      float t = logf(cv + sp_b(Bb[i * HS_N + j]) + DELTA_F);
      atomicAdd(&Eb[i * HS_N + j], t);
    }
  }
}

// ---------------- 8: fused upsample + log-lik + online softmax moments ----------
__global__ __launch_bounds__(256) void k_moments(const float* __restrict__ hm,
                                                 const float* __restrict__ energy128,
                                                 const float* __restrict__ st_mean,
                                                 const float* __restrict__ st_scale,
                                                 const float* __restrict__ beta,
                                                 float* __restrict__ out) {
  __shared__ float e128[HS_N * HS_N];  // 64 KB slice cache
  __shared__ float rm[256], rs[256], rx[256], ry[256];
  const int bl = blockIdx.x;
  const int l = bl % L_N;
  const int tid = threadIdx.x;
  const float* E = energy128 + (size_t)bl * (HS_N * HS_N);
  for (int i = tid; i < HS_N * HS_N; i += 256) e128[i] = E[i];
  __syncthreads();

  const float mn = st_mean[l], sc = st_scale[l], bt = beta[l];
  const float* H = hm + (size_t)bl * (HM_N * HM_N);

  float m = -3.0e38f, s = 0.f, sx = 0.f, sy = 0.f;
  for (int p = tid; p < HM_N * HM_N; p += 256) {
    int r = p >> 9, c = p & 511;
    // 4-tap cubic upsample 128->512 (separable, edge-renormalized)
    float yf = (float)r * 0.25f - 0.375f;
    float xf = (float)c * 0.25f - 0.375f;
    int iy = (int)floorf(yf), ix = (int)floorf(xf);
    float wy[4], wx[4], sumy = 0.f, sumx = 0.f;
#pragma unroll
    for (int u = 0; u < 4; ++u) {
      int ty = iy - 1 + u;
      float w = (ty >= 0 && ty < HS_N) ? cubic_w(yf - (float)ty) : 0.f;
      wy[u] = w; sumy += w;
      int tx = ix - 1 + u;
      float wc = (tx >= 0 && tx < HS_N) ? cubic_w(xf - (float)tx) : 0.f;
      wx[u] = wc; sumx += wc;
    }
    float up = 0.f;
#pragma unroll
    for (int u = 0; u < 4; ++u) {
      if (wy[u] == 0.f) continue;
      const float* er = &e128[(iy - 1 + u) << 7];
#pragma unroll
      for (int w = 0; w < 4; ++w)
        if (wx[w] != 0.f) up += wy[u] * wx[w] * er[ix - 1 + w];
    }
    up /= (sumy * sumx);
    float hn = (H[p] - mn) * sc + bt;
    float e = up + logf(sp_b(hn) + DELTA_F);
    if (e > m) {
      float f = expf(m - e);
      s *= f; sx *= f; sy *= f; m = e;
    }
    float w = expf(e - m);
    s += w; sx += w * (float)r; sy += w * (float)c;
  }
  rm[tid] = m; rs[tid] = s; rx[tid] = sx; ry[tid] = sy;
  __syncthreads();
  for (int st = 128; st; st >>= 1) {
    if (tid < st) {
      float m1 = rm[tid], m2 = rm[tid + st];
      float M = fmaxf(m1, m2);
      float f1 = expf(m1 - M), f2 = expf(m2 - M);
      rm[tid] = M;
      rs[tid] = rs[tid] * f1 + rs[tid + st] * f2;
      rx[tid] = rx[tid] * f1 + rx[tid + st] * f2;
      ry[tid] = ry[tid] * f1 + ry[tid + st] * f2;
    }
    __syncthreads();
  }
  if (tid == 0) {
    out[bl * 3 + 0] = 1.f;
    out[bl * 3 + 1] = rx[0] / rs[0];
    out[bl * 3 + 2] = ry[0] / rs[0];
  }
}

// ---------------- launcher ----------------
extern "C" void kernel_launch(void* const* d_in, const int* in_sizes, int n_in,
                              void* d_out, int out_size, void* d_ws, size_t ws_size,
                              hipStream_t stream) {
  (void)in_sizes; (void)n_in; (void)out_size; (void)ws_size;
  const float* heat  = (const float*)d_in[0];   // (2,37,512,512)
  const float* cond  = (const float*)d_in[1];   // (37,8,64,64)
  const float* bias  = (const float*)d_in[2];   // (37,8,128,128)
  const float* gamma = (const float*)d_in[3];   // (37)
  const float* beta  = (const float*)d_in[4];   // (37)
  const int*   vidx  = (const int*)d_in[5];     // (37,8)
  float* out = (float*)d_out;                   // (2,37,3)

  float* ws = (float*)d_ws;
  float* st_mean  = ws;                                   // 37
  float* st_scale = ws + 64;                              // 37
  int*   pair_l   = (int*)(ws + 128);                     // 296
  float* energy   = ws + 512;                             // 2*37*128*128
  float* Lpad     = energy + (size_t)2 * L_N * HS_N * HS_N;   // 2*37*256*256
  float* tmp1     = Lpad + (size_t)2 * L_N * 256 * 256;       // 2*37*512*128
  float* prior    = tmp1;  // reuse: prior written only after tmp1 is consumed

  const size_t nRows  = (size_t)2 * L_N * HM_N * HS_N;    // 4,849,664
  const size_t nCols  = (size_t)2 * L_N * 256 * 256;      // 4,849,664
  const size_t nPrior = (size_t)L_N * K_N * HS_N * HS_N;  // 4,849,664
  const size_t nE     = (size_t)2 * L_N * HS_N * HS_N;    // 1,212,416

  k_bn_stats<<<L_N, 256, 0, stream>>>(heat, gamma, st_mean, st_scale);
  k_pairs<<<1, L_N * K_N, 0, stream>>>(vidx, pair_l);
  k_resize_rows<<<(unsigned)((nRows + 255) / 256), 256, 0, stream>>>(heat, tmp1);
  k_resize_cols_pad<<<(unsigned)((nCols + 255) / 256), 256, 0, stream>>>(
      tmp1, st_mean, st_scale, beta, Lpad);
  k_prior<<<(unsigned)((nPrior + 255) / 256), 256, 0, stream>>>(cond, prior);
  k_zero<<<(unsigned)((nE + 255) / 256), 256, 0, stream>>>(energy, nE);
  k_conv_wmma<<<dim3(HS_N / 16, L_N, 2), 256, 0, stream>>>(
      Lpad, prior, bias, pair_l, energy);
  k_moments<<<2 * L_N, 256, 0, stream>>>(heat, energy, st_mean, st_scale, beta, out);
}